// IF_48464410968111
// MI455X (gfx1250) — compile-verified
//
#include <hip/hip_runtime.h>
#include <stdint.h>

// IF (integrate-and-fire) forward, T=8 timesteps, soft reset, spike = thr.
// x: (T*B, C, H, W) fp32 viewed as (T, B, C, H, W); per-t slice is contiguous.
// Memory-bound: 268 MB traffic -> ~11.5 us floor at 23.3 TB/s.
// CDNA5 path: async global->LDS B128 copies (ASYNCcnt) pipeline all 8
// timesteps per wave with zero VGPR pressure, ds_load_b128 readback,
// non-temporal B128 stores.

typedef float f4 __attribute__((ext_vector_type(4)));
typedef int v4i __attribute__((vector_size(16)));

#define T_STEPS 8
#define ELEMS_PER_T (32 * 512 * 16 * 16) /* 4,194,304 floats per timestep */
#define BLOCK_THREADS 256
#define FLOATS_PER_BLOCK (BLOCK_THREADS * 4) /* 1024 floats = 4 KiB per t */

#if defined(__HIP_DEVICE_COMPILE__)

typedef __attribute__((address_space(1))) v4i* global_v4i_ptr;
typedef __attribute__((address_space(3))) v4i* lds_v4i_ptr;
typedef __attribute__((address_space(3))) volatile f4* lds_f4_vptr;

template <int N>
__device__ __forceinline__ void wait_async() {
#if __has_builtin(__builtin_amdgcn_s_wait_asynccnt)
  __builtin_amdgcn_s_wait_asynccnt(N);
#else
  asm volatile("s_wait_asynccnt %0" ::"i"(N) : "memory");
#endif
}

__device__ __forceinline__ void async_copy_b128(const float* gptr, float* lptr) {
#if __has_builtin(__builtin_amdgcn_global_load_async_to_lds_b128)
  __builtin_amdgcn_global_load_async_to_lds_b128(
      (global_v4i_ptr)(void*)gptr, (lds_v4i_ptr)(void*)lptr,
      /*offset=*/0, /*cpol=*/0);
#else
  unsigned loff = (unsigned)(uintptr_t)lptr; // low 32 bits of generic LDS ptr = LDS offset
  asm volatile("global_load_async_to_lds_b128 %0, %1, off" ::"v"(loff),
               "v"((uint64_t)(uintptr_t)gptr)
               : "memory");
#endif
}

// AS(3)-qualified volatile read -> true ds_load_b128, ordered vs the
// side-effecting s_wait_asynccnt (volatile ops can't be reordered across it).
__device__ __forceinline__ f4 lds_read_b128(float* lptr) {
  return *(lds_f4_vptr)(void*)lptr;
}

#endif // __HIP_DEVICE_COMPILE__

__global__ __launch_bounds__(BLOCK_THREADS) void if_fwd_kernel(
    const float* __restrict__ x, const float* __restrict__ thr_p,
    float* __restrict__ out) {
#if defined(__HIP_DEVICE_COMPILE__)
  __shared__ float lds[T_STEPS][FLOATS_PER_BLOCK]; // 32 KiB

  const int tid = threadIdx.x;
  const int tid4 = tid * 4;
  const size_t base = (size_t)blockIdx.x * FLOATS_PER_BLOCK + (size_t)tid4;

  // Issue all 8 timesteps' copies up front: 8 async B128 ops in flight per
  // lane, tracked by ASYNCcnt, completing in issue order.
#pragma unroll
  for (int t = 0; t < T_STEPS; ++t) {
    async_copy_b128(x + (size_t)t * ELEMS_PER_T + base, &lds[t][tid4]);
  }

  const float thr = thr_p[0];     // scalar learnable threshold (overlaps with copies)
  const float m0 = 0.5f * thr;    // mem init = 0.5 * thresh
  f4 mem;
  mem[0] = m0; mem[1] = m0; mem[2] = m0; mem[3] = m0;

#define IF_STEP(t, w)                                                         \
  {                                                                           \
    wait_async<w>(); /* load t complete (in-order ASYNCcnt) */                \
    f4 xv = lds_read_b128(&lds[t][tid4]); /* ds_load_b128 */                  \
    f4 sp;                                                                    \
    _Pragma("unroll") for (int j = 0; j < 4; ++j) {                           \
      float m = mem[j] + xv[j];                                               \
      float s = (m >= thr) ? thr : 0.0f; /* spike * thresh */                 \
      sp[j] = s;                                                              \
      mem[j] = m - s; /* soft reset */                                        \
    }                                                                         \
    __builtin_nontemporal_store(                                              \
        sp, (f4*)(out + (size_t)(t) * ELEMS_PER_T + base));                   \
  }

  IF_STEP(0, 7)
  IF_STEP(1, 6)
  IF_STEP(2, 5)
  IF_STEP(3, 4)
  IF_STEP(4, 3)
  IF_STEP(5, 2)
  IF_STEP(6, 1)
  IF_STEP(7, 0)
#undef IF_STEP
#endif // __HIP_DEVICE_COMPILE__
}

extern "C" void kernel_launch(void* const* d_in, const int* in_sizes, int n_in,
                              void* d_out, int out_size, void* d_ws,
                              size_t ws_size, hipStream_t stream) {
  const float* x = (const float*)d_in[0];      // (256, 512, 16, 16) fp32
  const float* thr = (const float*)d_in[1];    // (1,) fp32
  float* out = (float*)d_out;                  // same shape as x

  const int per_t = in_sizes[0] / T_STEPS;          // 4,194,304
  const int blocks = per_t / FLOATS_PER_BLOCK;      // 4096 (exact)

  if_fwd_kernel<<<blocks, BLOCK_THREADS, 0, stream>>>(x, thr, out);

  (void)n_in;
  (void)out_size;
  (void)d_ws;
  (void)ws_size;
}